// SVD_30975304139306
// MI455X (gfx1250) — compile-verified
//
#include <hip/hip_runtime.h>

// Batched Kabsch/Procrustes (no reflection fix, matching the reference):
//   R = polar_ortho(H^T),  t = Ybar - Xbar R^T
// H accumulated via V_WMMA_F32_16X16X4_F32 with an augmented ones row/col so
// one WMMA chain produces Sxy (3x3), sum(X), sum(Y) and the point count.
//
// Memory-bound: reads src + pred_flow only (~196.6 MB -> ~8.4us HBM floor);
// gt_flow is a dead gate in the reference and is never touched.
//
// v2 changes vs v1 (from disasm feedback):
//  - zero-tail LDS slot + per-lane stride-0 indexing for padding lanes:
//    removes s_and_saveexec/exec-restore branches and per-iteration v_mov 0s
//  - dual accumulators (even/odd chunks) to break the WMMA RAW chain and
//    overlap ds_load latency with XDL execution

#define PPC    500   // points per cloud
#define PAIRS  250   // PPC/2
#define CHUNKS 125   // PPC/4  (K=4 points per WMMA)
#define ZSLOT  (PAIRS * 4)   // index of the zeroed padding float2

typedef float v2f __attribute__((ext_vector_type(2)));
typedef float v8f __attribute__((ext_vector_type(8)));

__global__ __launch_bounds__(32)
void SVD_30975304139306_kernel(const float* __restrict__ src,
                               const float* __restrict__ flow,
                               float* __restrict__ out,
                               int Btot)
{
    // Pair-interleaved staging: XP[q*4+c] = (X[2q][c], X[2q+1][c]), c=0..2,
    // slot c==3 holds (1,1) -> augmented ones row/col. Entry [ZSLOT] is the
    // shared zero pad read by lanes whose A/B rows are zero (c=4..15).
    __shared__ float2 XP[PAIRS * 4 + 1];
    __shared__ float2 YP[PAIRS * 4 + 1];
    __shared__ float  Hs[16];          // 4x4 augmented H_raw

    const int b = blockIdx.x;
    if (b >= Btot) return;
    const int lane = threadIdx.x;      // wave32: one wave per cloud

    const float* sB = src  + (size_t)b * (PPC * 3);
    const float* fB = flow + (size_t)b * (PPC * 3);

    if (lane == 0) {                   // zero pad slots (LDS is uninitialized)
        XP[ZSLOT] = make_float2(0.f, 0.f);
        YP[ZSLOT] = make_float2(0.f, 0.f);
    }

    // ---- Stage cloud into LDS (coalesced: 24 contiguous bytes per lane) ----
    for (int q = lane; q < PAIRS; q += 32) {
        const float* sp = sB + q * 6;
        const float* fp = fB + q * 6;
        float s0 = sp[0], s1 = sp[1], s2 = sp[2], s3 = sp[3], s4 = sp[4], s5 = sp[5];
        float g0 = fp[0], g1 = fp[1], g2 = fp[2], g3 = fp[3], g4 = fp[4], g5 = fp[5];
        XP[q * 4 + 0] = make_float2(s0, s3);
        XP[q * 4 + 1] = make_float2(s1, s4);
        XP[q * 4 + 2] = make_float2(s2, s5);
        XP[q * 4 + 3] = make_float2(1.f, 1.f);
        YP[q * 4 + 0] = make_float2(s0 + g0, s3 + g3);   // Y = X + flow
        YP[q * 4 + 1] = make_float2(s1 + g1, s4 + g4);
        YP[q * 4 + 2] = make_float2(s2 + g2, s5 + g5);
        YP[q * 4 + 3] = make_float2(1.f, 1.f);
    }
    __syncthreads();

    // ---- WMMA accumulation: D[m][n] += sum_k A[m][k]*B[k][n] -------------
    // f32 16x16x4 layout: lanes 0-15 hold K={0,1} in vgpr{0,1}; lanes 16-31
    // hold K={2,3}.  Chunk i uses pair q = 2i+hi, component c.
    // Element index for chunk i: 8*i + 4*hi + c (active lanes), ZSLOT else.
    const int c   = lane & 15;
    const int hi  = lane >> 4;
    const bool act = (c < 4);
    int       idx = act ? (4 * hi + c) : ZSLOT;   // float2 index
    const int st  = act ? 8 : 0;                  // per-chunk stride (0 = stay on pad)

    v8f acc0 = {};
    v8f acc1 = {};
    for (int i = 0; i < (CHUNKS - 1) / 2; ++i) {  // 62 double-chunks
        v2f a0 = *(const v2f*)&XP[idx];
        v2f b0 = *(const v2f*)&YP[idx];
        v2f a1 = *(const v2f*)&XP[idx + st];
        v2f b1 = *(const v2f*)&YP[idx + st];
        acc0 = __builtin_amdgcn_wmma_f32_16x16x4_f32(false, a0, false, b0,
                                                     (short)0, acc0, false, false);
        acc1 = __builtin_amdgcn_wmma_f32_16x16x4_f32(false, a1, false, b1,
                                                     (short)0, acc1, false, false);
        idx += 2 * st;
    }
    {   // tail chunk (125 = 62*2 + 1)
        v2f a0 = *(const v2f*)&XP[idx];
        v2f b0 = *(const v2f*)&YP[idx];
        acc0 = __builtin_amdgcn_wmma_f32_16x16x4_f32(false, a0, false, b0,
                                                     (short)0, acc0, false, false);
    }

    // D layout: vgpr r, lanes 0-15 -> M=r, N=lane.  Need the 4x4 corner.
    if (lane < 4) {
        Hs[0 * 4 + lane] = acc0[0] + acc1[0];
        Hs[1 * 4 + lane] = acc0[1] + acc1[1];
        Hs[2 * 4 + lane] = acc0[2] + acc1[2];
        Hs[3 * 4 + lane] = acc0[3] + acc1[3];
    }
    __syncthreads();

    // ---- 3x3 polar decomposition + outputs (lane 0) ----------------------
    if (lane == 0) {
        const float sx0 = Hs[3],  sx1 = Hs[7],  sx2 = Hs[11];   // sum(X)
        const float sy0 = Hs[12], sy1 = Hs[13], sy2 = Hs[14];   // sum(Y)
        const float inv = 1.0f / Hs[15];                        // 1/P
        const float xb0 = sx0 * inv, xb1 = sx1 * inv, xb2 = sx2 * inv;
        const float yb0 = sy0 * inv, yb1 = sy1 * inv, yb2 = sy2 * inv;

        // M = H^T, H[i][j] = Sxy[i][j] - sx_i*sy_j/P   (row-major M[i*3+j])
        float m[9];
        m[0] = Hs[0]  - sx0 * sy0 * inv;
        m[1] = Hs[4]  - sx1 * sy0 * inv;
        m[2] = Hs[8]  - sx2 * sy0 * inv;
        m[3] = Hs[1]  - sx0 * sy1 * inv;
        m[4] = Hs[5]  - sx1 * sy1 * inv;
        m[5] = Hs[9]  - sx2 * sy1 * inv;
        m[6] = Hs[2]  - sx0 * sy2 * inv;
        m[7] = Hs[6]  - sx1 * sy2 * inv;
        m[8] = Hs[10] - sx2 * sy2 * inv;

        // Scaled Newton polar iteration: X <- 0.5*(g*X + (1/g)*X^{-T}),
        // g = (||X^{-1}||_F / ||X||_F)^{1/2}.  Quadratic convergence.
        #pragma unroll
        for (int it = 0; it < 8; ++it) {
            const float c00 =  (m[4] * m[8] - m[5] * m[7]);
            const float c01 = -(m[3] * m[8] - m[5] * m[6]);
            const float c02 =  (m[3] * m[7] - m[4] * m[6]);
            const float c10 = -(m[1] * m[8] - m[2] * m[7]);
            const float c11 =  (m[0] * m[8] - m[2] * m[6]);
            const float c12 = -(m[0] * m[7] - m[1] * m[6]);
            const float c20 =  (m[1] * m[5] - m[2] * m[4]);
            const float c21 = -(m[0] * m[5] - m[2] * m[3]);
            const float c22 =  (m[0] * m[4] - m[1] * m[3]);
            const float det = m[0] * c00 + m[1] * c01 + m[2] * c02;
            if (fabsf(det) < 1e-30f) break;           // degenerate guard
            const float id = 1.0f / det;              // X^{-T} = cof * id
            const float i0 = c00 * id, i1 = c01 * id, i2 = c02 * id;
            const float i3 = c10 * id, i4 = c11 * id, i5 = c12 * id;
            const float i6 = c20 * id, i7 = c21 * id, i8 = c22 * id;
            const float nA = m[0]*m[0]+m[1]*m[1]+m[2]*m[2]+m[3]*m[3]+m[4]*m[4]
                           + m[5]*m[5]+m[6]*m[6]+m[7]*m[7]+m[8]*m[8];
            const float nI = i0*i0+i1*i1+i2*i2+i3*i3+i4*i4+i5*i5+i6*i6+i7*i7+i8*i8;
            const float g  = sqrtf(sqrtf(nI / nA));
            const float ga = 0.5f * g;
            const float gi = 0.5f / g;
            m[0] = ga * m[0] + gi * i0;  m[1] = ga * m[1] + gi * i1;
            m[2] = ga * m[2] + gi * i2;  m[3] = ga * m[3] + gi * i3;
            m[4] = ga * m[4] + gi * i4;  m[5] = ga * m[5] + gi * i5;
            m[6] = ga * m[6] + gi * i6;  m[7] = ga * m[7] + gi * i7;
            m[8] = ga * m[8] + gi * i8;
        }

        // R = m (row-major).  Write R then t = Ybar - Xbar . R_rowj
        float* Ro = out + (size_t)b * 9;
        #pragma unroll
        for (int k = 0; k < 9; ++k) Ro[k] = m[k];

        float* To = out + (size_t)Btot * 9 + (size_t)b * 3;
        To[0] = yb0 - (xb0 * m[0] + xb1 * m[1] + xb2 * m[2]);
        To[1] = yb1 - (xb0 * m[3] + xb1 * m[4] + xb2 * m[5]);
        To[2] = yb2 - (xb0 * m[6] + xb1 * m[7] + xb2 * m[8]);
    }
}

extern "C" void kernel_launch(void* const* d_in, const int* in_sizes, int n_in,
                              void* d_out, int out_size, void* d_ws, size_t ws_size,
                              hipStream_t stream) {
    const float* src  = (const float*)d_in[0];   // [B*500, 3]
    const float* flow = (const float*)d_in[1];   // [B*500, 3] (pred_flow)
    // d_in[2] (gt_flow) is a dead gate in the reference — intentionally unread.
    float* out = (float*)d_out;                  // [B*9] R ++ [B*3] t
    (void)d_ws; (void)ws_size; (void)n_in; (void)out_size;

    const int Btot = in_sizes[0] / (PPC * 3);    // 16384 for the given setup
    SVD_30975304139306_kernel<<<dim3(Btot), dim3(32), 0, stream>>>(src, flow, out, Btot);
}